// HeteroRGCNLayer_13280038879653
// MI455X (gfx1250) — compile-verified
//
#include <hip/hip_runtime.h>
#include <hip/hip_bf16.h>

#define DIM 128

typedef float v2f __attribute__((ext_vector_type(2)));
typedef float v8f __attribute__((ext_vector_type(8)));

// out[n,d] = (1/R) * sum_r b[r,d]   (bias mean; acc starts here)
__global__ __launch_bounds__(256) void init_out_kernel(const float* __restrict__ b,
                                                       float* __restrict__ out,
                                                       int total, int R) {
  int i = blockIdx.x * blockDim.x + threadIdx.x;
  if (i >= total) return;
  int d = i & (DIM - 1);
  float acc = 0.0f;
  for (int r = 0; r < R; ++r) acc += b[r * DIM + d];
  out[i] = acc / (float)R;
}

// per-edge degree counts (float atomics; exact for counts < 2^24)
__global__ __launch_bounds__(256) void degree_kernel(const int* __restrict__ src,
                                                     const int* __restrict__ dst,
                                                     float* __restrict__ deg_out,
                                                     float* __restrict__ deg_in,
                                                     int E) {
  int e = blockIdx.x * blockDim.x + threadIdx.x;
  if (e >= E) return;
  atomicAdd(&deg_out[src[e]], 1.0f);
  atomicAdd(&deg_in[dst[e]], 1.0f);
}

// one wave per edge: agg[dst] += x[src] * ew * rsqrt(max(deg_out[src],1))
__global__ __launch_bounds__(256) void scatter_kernel(const float* __restrict__ x,
                                                      const int* __restrict__ src,
                                                      const int* __restrict__ dst,
                                                      const float* __restrict__ ew,
                                                      const float* __restrict__ deg_out,
                                                      float* __restrict__ agg,
                                                      int E) {
  int wave = (int)((blockIdx.x * blockDim.x + threadIdx.x) >> 5);
  int lane = threadIdx.x & 31;
  if (wave >= E) return;
  int s = src[wave];
  int t = dst[wave];
  float w = ew[wave] * rsqrtf(fmaxf(deg_out[s], 1.0f));
  const float4* xs = (const float4*)(x + (size_t)s * DIM);
  float4 v = xs[lane];                        // lane covers 4 of 128 columns
  float* ap = agg + (size_t)t * DIM + lane * 4;
  atomicAdd(ap + 0, v.x * w);
  atomicAdd(ap + 1, v.y * w);
  atomicAdd(ap + 2, v.z * w);
  atomicAdd(ap + 3, v.w * w);
}

// out += inv_r * (diag(rsqrt(max(deg_in,1))) * agg) @ Wr
// Block = 256 threads = 8 waves; block covers 16 rows x 128 cols.
// Wave w computes the 16x16 tile at columns [16w, 16w+16) via V_WMMA_F32_16X16X4_F32.
__global__ __launch_bounds__(256) void gemm_acc_kernel(const float* __restrict__ agg,
                                                       const float* __restrict__ Wr,
                                                       const float* __restrict__ deg_in,
                                                       float* __restrict__ out,
                                                       int nrows, float inv_r) {
  const int lane = threadIdx.x & 31;
  const int wv   = threadIdx.x >> 5;        // 0..7 -> 16-col block
  const int m0   = blockIdx.x * 16;
  const int col0 = wv * 16;
  const int half = lane >> 4;               // lane 16..31 handle K+2 / M+8
  const int l    = lane & 15;

  int arow = m0 + l;
  if (arow >= nrows) arow = nrows - 1;      // clamp loads; N%16==0 in practice
  const float ascale = rsqrtf(fmaxf(deg_in[arow], 1.0f));
  const float* aptr = agg + (size_t)arow * DIM + half * 2;     // A: M=l(+0), K base half*2
  const float* bptr = Wr + (size_t)(half * 2) * DIM + col0 + l; // B: K base half*2, N=l

  v8f c = {};
#pragma unroll
  for (int k0 = 0; k0 < DIM; k0 += 4) {
    v2f a, b;
    a.x = aptr[k0 + 0] * ascale;            // A[m][k0+half*2]
    a.y = aptr[k0 + 1] * ascale;            // A[m][k0+half*2+1]
    b.x = bptr[(size_t)(k0 + 0) * DIM];     // W[k0+half*2][col0+l]
    b.y = bptr[(size_t)(k0 + 1) * DIM];     // W[k0+half*2+1][col0+l]
    // 8 args: (neg_a, A, neg_b, B, c_mod, C, reuse_a, reuse_b)
    c = __builtin_amdgcn_wmma_f32_16x16x4_f32(false, a, false, b, (short)0, c,
                                              false, false);
  }

  // C/D layout: VGPR j -> row m0 + j + half*8, col col0 + l
#pragma unroll
  for (int j = 0; j < 8; ++j) {
    int row = m0 + j + half * 8;
    if (row < nrows)
      out[(size_t)row * DIM + col0 + l] += c[j] * inv_r;
  }
}

extern "C" void kernel_launch(void* const* d_in, const int* in_sizes, int n_in,
                              void* d_out, int out_size, void* d_ws, size_t ws_size,
                              hipStream_t stream) {
  const float* x   = (const float*)d_in[0];   // [N,128]
  const float* W   = (const float*)d_in[1];   // [R,128,128]
  const float* b   = (const float*)d_in[2];   // [R,128]
  const int*   src = (const int*)d_in[3];     // [R,E]
  const int*   dst = (const int*)d_in[4];     // [R,E]
  const float* ew  = (const float*)d_in[5];   // [R,E]
  float* out = (float*)d_out;

  const int N = in_sizes[0] / DIM;
  const int R = in_sizes[2] / DIM;
  const int E = in_sizes[3] / R;
  const float inv_r = 1.0f / (float)R;

  // workspace: [deg_out N][deg_in N][agg N*128], zeroed fresh per relation
  float* deg_out_p = (float*)d_ws;
  float* deg_in_p  = deg_out_p + N;
  float* agg       = deg_in_p + N;
  const size_t zero_bytes = ((size_t)2 * N + (size_t)N * DIM) * sizeof(float);
  if (ws_size < zero_bytes) return;

  const int total = N * DIM;
  init_out_kernel<<<(total + 255) / 256, 256, 0, stream>>>(b, out, total, R);

  for (int r = 0; r < R; ++r) {
    hipMemsetAsync(d_ws, 0, zero_bytes, stream);

    const int* sr = src + (size_t)r * E;
    const int* dr = dst + (size_t)r * E;
    const float* er = ew + (size_t)r * E;
    const float* Wr = W + (size_t)r * DIM * DIM;

    degree_kernel<<<(E + 255) / 256, 256, 0, stream>>>(sr, dr, deg_out_p, deg_in_p, E);

    // one wave (32 lanes) per edge
    scatter_kernel<<<(E + 7) / 8, 256, 0, stream>>>(x, sr, dr, er, deg_out_p, agg, E);

    gemm_acc_kernel<<<(N + 15) / 16, 256, 0, stream>>>(agg, Wr, deg_in_p, out, N, inv_r);
  }
}